// ORAlign1d_17952963297816
// MI455X (gfx1250) — compile-verified
//
#include <hip/hip_runtime.h>

// ORAlign1d for MI455X (gfx1250, wave32).
// x: [N, F*O] fp32, O=8. Per group of 8 contiguous floats:
//   m = argmax (first occurrence), out[o] = in[(o+m) & 7].
// Streaming kernel: 512 MB total @ 23.3 TB/s => ~22 us floor.
// VALU-minimized: every condition is a raw float compare consumed directly
// as a cndmask selector; rotate-by-4 folded into store addressing; 32-bit
// offsets so loads/stores use the saddr (SGPR base + VGPR offset) form.

typedef float floatx4 __attribute__((ext_vector_type(4)));

// Process one group (lo = v[0..3], hi = v[4..7]): emits the two 16-byte
// store operands A (goes to +sw) and B (goes to +(sw^16)), where sw folds
// the rotate-by-4 into the store addresses.
__device__ __forceinline__ void align_group(const floatx4 lo, const floatx4 hi,
                                            floatx4& A, floatx4& B,
                                            unsigned& sw) {
    float v[8];
#pragma unroll
    for (int i = 0; i < 4; ++i) { v[i] = lo[i]; v[i + 4] = hi[i]; }

    // Tournament: strict > at every level => lower index wins ties,
    // matching jnp.argmax first-occurrence semantics.
    const bool  b0  = v[1] > v[0];
    const bool  b1  = v[3] > v[2];
    const bool  b2  = v[5] > v[4];
    const bool  b3  = v[7] > v[6];
    const float w01 = b0 ? v[1] : v[0];
    const float w23 = b1 ? v[3] : v[2];
    const float w45 = b2 ? v[5] : v[4];
    const float w67 = b3 ? v[7] : v[6];
    const bool  c0  = w23 > w01;
    const bool  c1  = w67 > w45;
    const float ha  = c0 ? w23 : w01;
    const float hb  = c1 ? w67 : w45;
    const bool  r4  = hb > ha;           // tie -> lower half
    const float pa  = r4 ? w45 : w01;
    const float pb  = r4 ? w67 : w23;
    const bool  r2  = pb > pa;           // tie -> lower pair
    const float f0  = r4 ? v[4] : v[0];
    const float f1  = r4 ? v[5] : v[1];
    const float f2  = r4 ? v[6] : v[2];
    const float f3  = r4 ? v[7] : v[3];
    const float e0  = r2 ? f2 : f0;
    const float e1  = r2 ? f3 : f1;
    const bool  r1  = e1 > e0;           // tie -> first element

    // Rotate-by-1 under r1 (8 cnd).
    float t1[8];
#pragma unroll
    for (int i = 0; i < 8; ++i) t1[i] = r1 ? v[(i + 1) & 7] : v[i];

    // Rotate-by-2 under r2, written directly into the store operands (8 cnd).
#pragma unroll
    for (int i = 0; i < 4; ++i) A[i] = r2 ? t1[(i + 2) & 7] : t1[i];
#pragma unroll
    for (int i = 0; i < 4; ++i) B[i] = r2 ? t1[(i + 6) & 7] : t1[i + 4];

    // Rotate-by-4 folded into the store addresses.
    sw = r4 ? 16u : 0u;
}

__global__ __launch_bounds__(256) void oralign8_kernel(
    const float* __restrict__ x, float* __restrict__ y, unsigned int nthreads) {
    const unsigned int t = blockIdx.x * 256u + threadIdx.x;
    if (t >= nthreads) return;

    // 32-bit byte offset (t < 2^22 so t<<6 < 2^28): enables saddr-form
    // global loads/stores (SGPR base + 32-bit VGPR offset), no 64-bit
    // address arithmetic in VALU.
    const unsigned int voff = t << 6;  // 16 floats (2 groups) per thread
    const char* __restrict__ ib = (const char*)x + voff;
    char* __restrict__       ob = (char*)y + voff;

    // Warm L2 for the next block's tile (16 KB ahead) -> global_prefetch_b8.
    __builtin_prefetch(ib + 16384, /*rw=*/0, /*locality=*/1);

    // 4 outstanding non-temporal b128 loads for memory-level parallelism.
    const floatx4 a = __builtin_nontemporal_load((const floatx4*)(ib));
    const floatx4 b = __builtin_nontemporal_load((const floatx4*)(ib + 16));
    const floatx4 c = __builtin_nontemporal_load((const floatx4*)(ib + 32));
    const floatx4 d = __builtin_nontemporal_load((const floatx4*)(ib + 48));

    floatx4 A0, B0, A1, B1;
    unsigned sw0, sw1;
    align_group(a, b, A0, B0, sw0);
    align_group(c, d, A1, B1, sw1);

    __builtin_nontemporal_store(A0, (floatx4*)(ob + sw0));
    __builtin_nontemporal_store(B0, (floatx4*)(ob + (sw0 ^ 16u)));
    __builtin_nontemporal_store(A1, (floatx4*)(ob + 32u + sw1));
    __builtin_nontemporal_store(B1, (floatx4*)(ob + 32u + (sw1 ^ 16u)));
}

extern "C" void kernel_launch(void* const* d_in, const int* in_sizes, int n_in,
                              void* d_out, int out_size, void* d_ws, size_t ws_size,
                              hipStream_t stream) {
    const float* x = (const float*)d_in[0];
    float*       y = (float*)d_out;

    // in_sizes[0] = N*C = 67,108,864 elements; O fixed at 8 by the reference.
    const unsigned int n        = (unsigned int)in_sizes[0];
    const unsigned int nthreads = n / 16u;  // 16 floats (2 groups) per thread
    const unsigned int block    = 256u;
    const unsigned int grid     = (nthreads + block - 1u) / block;

    oralign8_kernel<<<dim3(grid), dim3(block), 0, stream>>>(x, y, nthreads);
}